// FNO3d_56942676410444
// MI455X (gfx1250) — compile-verified
//
#include <hip/hip_runtime.h>
#include <math.h>

// ---------------- problem constants ----------------
#define BN   8
#define NX   64
#define NY   64
#define NT   40
#define WCH  32          // network width
#define KT   8           // kept T modes (M3)
#define KXM  24          // kept X modes (2*M1)
#define KYM  24          // kept Y modes (2*M2)
#define PTS  (BN*NX*NY*NT)   // 1,310,720 points
#define LINES (BN*NX*NY)     // 32,768 (b,x,y) lines
#define PI_F 3.14159265358979323846f

typedef float v2f __attribute__((ext_vector_type(2)));
typedef float v8f __attribute__((ext_vector_type(8)));

__device__ __forceinline__ float gelu_exact(float x) {
    return 0.5f * x * (1.0f + erff(x * 0.70710678118654752f));
}

// ---------------- fc0: lift 5 -> 32, channels-last ----------------
__global__ void k_fc0(const float* __restrict__ h, const float* __restrict__ xg,
                      const float* __restrict__ w, const float* __restrict__ b,
                      float* __restrict__ v) {
    size_t idx = (size_t)blockIdx.x * blockDim.x + threadIdx.x;  // PTS*WCH threads
    size_t p = idx >> 5;
    int c = (int)(idx & 31);
    float acc = b[c];
    acc += h[p*2 + 0] * w[0*WCH + c];
    acc += h[p*2 + 1] * w[1*WCH + c];
    acc += xg[p*3 + 0] * w[2*WCH + c];
    acc += xg[p*3 + 1] * w[3*WCH + c];
    acc += xg[p*3 + 2] * w[4*WCH + c];
    v[idx] = acc;
}

// ---------------- forward DFT along T: 40 real -> 8 complex ----------------
__global__ void k_dft_t(const float* __restrict__ v, float2* __restrict__ Vt) {
    int L = blockIdx.x;                  // (b,x,y) line
    int tid = threadIdx.x;               // 256 = 8 k3 x 32 c
    int k3 = tid >> 5, c = tid & 31;
    __shared__ float2 tw[KT * NT];
    for (int i = tid; i < KT * NT; i += 256) {
        int k = i / NT, t = i % NT;
        int ph = (k * t) % NT;
        float sn, cs;
        __sincosf(-2.0f * PI_F * (float)ph / (float)NT, &sn, &cs);
        tw[i] = make_float2(cs, sn);
    }
    __syncthreads();
    const float* src = v + (size_t)L * NT * WCH + c;
    float re = 0.f, im = 0.f;
    for (int t = 0; t < NT; ++t) {
        float a = src[(size_t)t * WCH];
        float2 w = tw[k3 * NT + t];
        re += a * w.x; im += a * w.y;
    }
    Vt[((size_t)L * KT + k3) * WCH + c] = make_float2(re, im);
}

// ---------------- forward DFT along Y: 64 -> 24 modes ----------------
__global__ void k_dft_y(const float2* __restrict__ Vt, float2* __restrict__ Vy) {
    int bidx = blockIdx.x;               // (b*NX + x)*KYM + ky
    int ky = bidx % KYM;
    int bx = bidx / KYM;
    int tid = threadIdx.x;
    int k3 = tid >> 5, c = tid & 31;
    __shared__ float2 tw[NY];
    if (tid < NY) {
        int fy = (ky < 12) ? ky : (ky + 40);
        int ph = (fy * tid) & 63;
        float sn, cs;
        __sincosf(-2.0f * PI_F * (float)ph / 64.0f, &sn, &cs);
        tw[tid] = make_float2(cs, sn);
    }
    __syncthreads();
    float re = 0.f, im = 0.f;
    for (int y = 0; y < NY; ++y) {
        float2 a = Vt[(((size_t)bx * NY + y) * KT + k3) * WCH + c];
        float2 w = tw[y];
        re += a.x * w.x - a.y * w.y;
        im += a.x * w.y + a.y * w.x;
    }
    Vy[(((size_t)bx * KYM + ky) * KT + k3) * WCH + c] = make_float2(re, im);
}

// ---------------- forward DFT along X: 64 -> 24 modes ----------------
__global__ void k_dft_x(const float2* __restrict__ Vy, float2* __restrict__ Vx) {
    int bidx = blockIdx.x;               // (b*KXM + kx)*KYM + ky
    int ky = bidx % KYM;
    int kx = (bidx / KYM) % KXM;
    int b  = bidx / (KXM * KYM);
    int tid = threadIdx.x;
    int k3 = tid >> 5, c = tid & 31;
    __shared__ float2 tw[NX];
    if (tid < NX) {
        int fx = (kx < 12) ? kx : (kx + 40);
        int ph = (fx * tid) & 63;
        float sn, cs;
        __sincosf(-2.0f * PI_F * (float)ph / 64.0f, &sn, &cs);
        tw[tid] = make_float2(cs, sn);
    }
    __syncthreads();
    float re = 0.f, im = 0.f;
    for (int x = 0; x < NX; ++x) {
        float2 a = Vy[((((size_t)b * NX + x) * KYM + ky) * KT + k3) * WCH + c];
        float2 w = tw[x];
        re += a.x * w.x - a.y * w.y;
        im += a.x * w.y + a.y * w.x;
    }
    Vx[((((size_t)b * KXM + kx) * KYM + ky) * KT + k3) * WCH + c] = make_float2(re, im);
}

// ---------------- repack spectral weights -> [corner][m1][m2][k3][i][o] float2 ----------------
__global__ void k_repack(const float* __restrict__ wr, const float* __restrict__ wi,
                         float2* __restrict__ Wc) {
    size_t id = (size_t)blockIdx.x * blockDim.x + threadIdx.x;  // 4*12*12*8*32*32
    int o = (int)(id & 31);
    size_t r = id >> 5;
    int i = (int)(r & 31); r >>= 5;
    int k3 = (int)(r & 7); r >>= 3;
    int m2 = (int)(r % 12); r /= 12;
    int m1 = (int)(r % 12);
    int corner = (int)(r / 12);
    size_t src = ((((size_t)corner * WCH + i) * WCH + o) * 12 + m1) * 12 + m2;
    src = src * 8 + k3;
    size_t dst = (((((size_t)corner * 12 + m1) * 12 + m2) * 8 + k3) * WCH + i) * WCH + o;
    Wc[dst] = make_float2(wr[src], wi[src]);
}

// ---------------- spectral channel mixing (complex 32x32 per mode) ----------------
__global__ void k_specmul(const float2* __restrict__ Vx, const float2* __restrict__ Wc,
                          float2* __restrict__ Vm) {
    int bidx = blockIdx.x;               // (b*KXM + kx)*KYM + ky
    int ky = bidx % KYM;
    int kx = (bidx / KYM) % KXM;
    int corner = ((kx >= 12) ? 1 : 0) + ((ky >= 12) ? 2 : 0);
    int m1 = (kx >= 12) ? (kx - 12) : kx;
    int m2 = (ky >= 12) ? (ky - 12) : ky;
    int tid = threadIdx.x;
    int k3 = tid >> 5, o = tid & 31;
    __shared__ float2 tileA[KT * WCH];
    size_t base = (size_t)bidx * KT * WCH;
    tileA[tid] = Vx[base + tid];
    __syncthreads();
    size_t wb = ((((size_t)corner * 12 + m1) * 12 + m2) * 8 + k3) * WCH * WCH + o;
    float re = 0.f, im = 0.f;
    for (int i = 0; i < WCH; ++i) {
        float2 a = tileA[k3 * WCH + i];
        float2 w = Wc[wb + (size_t)i * WCH];
        re += a.x * w.x - a.y * w.y;
        im += a.x * w.y + a.y * w.x;
    }
    Vm[base + tid] = make_float2(re, im);
}

// ---------------- inverse DFT along X: 24 modes -> 64 ----------------
__global__ void k_idft_x(const float2* __restrict__ Vm, float2* __restrict__ Wx) {
    int bidx = blockIdx.x;               // (b*NX + x)*KYM + ky
    int ky = bidx % KYM;
    int bx = bidx / KYM;
    int x = bx % NX;
    int b = bx / NX;
    int tid = threadIdx.x;
    int k3 = tid >> 5, o = tid & 31;
    __shared__ float2 tw[KXM];
    if (tid < KXM) {
        int fx = (tid < 12) ? tid : (tid + 40);
        int ph = (fx * x) & 63;
        float sn, cs;
        __sincosf(2.0f * PI_F * (float)ph / 64.0f, &sn, &cs);
        tw[tid] = make_float2(cs, sn);
    }
    __syncthreads();
    float re = 0.f, im = 0.f;
    for (int kx = 0; kx < KXM; ++kx) {
        float2 a = Vm[((((size_t)b * KXM + kx) * KYM + ky) * KT + k3) * WCH + o];
        float2 w = tw[kx];
        re += a.x * w.x - a.y * w.y;
        im += a.x * w.y + a.y * w.x;
    }
    Wx[(((size_t)bx * KYM + ky) * KT + k3) * WCH + o] = make_float2(re, im);
}

// ---------------- inverse DFT along Y: 24 modes -> 64 ----------------
__global__ void k_idft_y(const float2* __restrict__ Wx, float2* __restrict__ Wy) {
    int L = blockIdx.x;                  // (b*NX + x)*NY + y
    int y = L % NY;
    int bx = L / NY;
    int tid = threadIdx.x;
    int k3 = tid >> 5, o = tid & 31;
    __shared__ float2 tw[KYM];
    if (tid < KYM) {
        int fy = (tid < 12) ? tid : (tid + 40);
        int ph = (fy * y) & 63;
        float sn, cs;
        __sincosf(2.0f * PI_F * (float)ph / 64.0f, &sn, &cs);
        tw[tid] = make_float2(cs, sn);
    }
    __syncthreads();
    float re = 0.f, im = 0.f;
    for (int ky = 0; ky < KYM; ++ky) {
        float2 a = Wx[(((size_t)bx * KYM + ky) * KT + k3) * WCH + o];
        float2 w = tw[ky];
        re += a.x * w.x - a.y * w.y;
        im += a.x * w.y + a.y * w.x;
    }
    Wy[((size_t)L * KT + k3) * WCH + o] = make_float2(re, im);
}

// ---------------- conv1x1 (32->32) via V_WMMA_F32_16X16X4_F32 ----------------
__global__ void k_conv1x1(const float* __restrict__ v, const float* __restrict__ w,
                          const float* __restrict__ bias, float* __restrict__ out,
                          int ntiles) {
    int lane = threadIdx.x & 31;
    int wave = blockIdx.x * (blockDim.x >> 5) + (threadIdx.x >> 5);
    int nwaves = gridDim.x * (blockDim.x >> 5);
    int m = lane & 15;
    int kb = (lane >> 4) * 2;
    // B = w^T tiles: B[k][n] = w[n][k]; lane holds (k=kc*4+kb, k+1) at col n=m of N-tile nt
    v2f Bt[2][8];
    for (int nt = 0; nt < 2; ++nt)
        for (int kc = 0; kc < 8; ++kc) {
            const float* wp = w + (size_t)(nt * 16 + m) * WCH + kc * 4 + kb;
            Bt[nt][kc] = (v2f){wp[0], wp[1]};
        }
    float bv0 = bias[m], bv1 = bias[16 + m];
    for (int tile = wave; tile < ntiles; tile += nwaves) {
        size_t p0 = (size_t)tile * 16;
        v8f acc0 = {0.f,0.f,0.f,0.f,0.f,0.f,0.f,0.f};
        v8f acc1 = {0.f,0.f,0.f,0.f,0.f,0.f,0.f,0.f};
        for (int kc = 0; kc < 8; ++kc) {
            const float* ap = v + (p0 + m) * WCH + kc * 4 + kb;
            v2f A = (v2f){ap[0], ap[1]};
            acc0 = __builtin_amdgcn_wmma_f32_16x16x4_f32(false, A, false, Bt[0][kc],
                                                         (short)0, acc0, false, false);
            acc1 = __builtin_amdgcn_wmma_f32_16x16x4_f32(false, A, false, Bt[1][kc],
                                                         (short)0, acc1, false, false);
        }
        for (int r = 0; r < 8; ++r) {
            int row = r + (lane >> 4) * 8;
            float* op = out + (p0 + row) * WCH;
            op[m]      = acc0[r] + bv0;
            op[16 + m] = acc1[r] + bv1;
        }
    }
}

// ---------------- irfft along T (8 modes -> 40) + conv skip + optional GELU ----------------
__global__ void k_combine(const float2* __restrict__ Wy, const float* __restrict__ cb,
                          float* __restrict__ vout, int doAct) {
    int L = blockIdx.x;
    int tid = threadIdx.x;
    int o = tid & 31;
    int trow = tid >> 5;                 // 0..7
    const float2* F = Wy + (size_t)L * KT * WCH + o;
    float2 f[KT];
    for (int k = 0; k < KT; ++k) f[k] = F[(size_t)k * WCH];
    const float inv = 1.0f / ((float)NX * (float)NY * (float)NT);
    size_t pbase = (size_t)L * NT;
    for (int it = 0; it < 5; ++it) {
        int t = trow + it * 8;
        float s = f[0].x;
        for (int k = 1; k < KT; ++k) {
            int ph = (k * t) % NT;
            float sn, cs;
            __sincosf(2.0f * PI_F * (float)ph / (float)NT, &sn, &cs);
            s += 2.0f * (f[k].x * cs - f[k].y * sn);
        }
        float val = s * inv + cb[(pbase + t) * WCH + o];
        vout[(pbase + t) * WCH + o] = doAct ? gelu_exact(val) : val;
    }
}

// ---------------- fused fc1 (32->64, WMMA) + GELU + fc2 (64->2, shuffle reduce) ----------------
__global__ void k_fc12(const float* __restrict__ v, const float* __restrict__ w1,
                       const float* __restrict__ b1, const float* __restrict__ w2,
                       const float* __restrict__ b2, float* __restrict__ out, int ntiles) {
    int lane = threadIdx.x & 31;
    int wave = blockIdx.x * (blockDim.x >> 5) + (threadIdx.x >> 5);
    int nwaves = gridDim.x * (blockDim.x >> 5);
    int m = lane & 15;
    int hi = lane >> 4;
    int kb = hi * 2;
    // B tiles: B[k][n] = w1[k][n], n = nt*16 + m
    v2f Bt[4][8];
    for (int nt = 0; nt < 4; ++nt)
        for (int kc = 0; kc < 8; ++kc) {
            int k = kc * 4 + kb;
            int col = nt * 16 + m;
            Bt[nt][kc] = (v2f){w1[(size_t)k * 64 + col], w1[(size_t)(k + 1) * 64 + col]};
        }
    float b1v[4]; v2f w2v[4];
    for (int nt = 0; nt < 4; ++nt) {
        int col = nt * 16 + m;
        b1v[nt] = b1[col];
        w2v[nt] = (v2f){w2[(size_t)col * 2], w2[(size_t)col * 2 + 1]};
    }
    float b20 = b2[0], b21 = b2[1];
    for (int tile = wave; tile < ntiles; tile += nwaves) {
        size_t p0 = (size_t)tile * 16;
        v8f acc[4];
        for (int nt = 0; nt < 4; ++nt) acc[nt] = (v8f){0.f,0.f,0.f,0.f,0.f,0.f,0.f,0.f};
        for (int kc = 0; kc < 8; ++kc) {
            const float* ap = v + (p0 + m) * WCH + kc * 4 + kb;
            v2f A = (v2f){ap[0], ap[1]};
            for (int nt = 0; nt < 4; ++nt)
                acc[nt] = __builtin_amdgcn_wmma_f32_16x16x4_f32(false, A, false, Bt[nt][kc],
                                                                (short)0, acc[nt], false, false);
        }
        // GELU + fc2 partials over this lane's 4 columns, rows r(+hi*8)
        float p0a[8], p1a[8];
        for (int r = 0; r < 8; ++r) {
            float s0 = 0.f, s1 = 0.f;
            for (int nt = 0; nt < 4; ++nt) {
                float g = gelu_exact(acc[nt][r] + b1v[nt]);
                s0 += g * w2v[nt].x;
                s1 += g * w2v[nt].y;
            }
            p0a[r] = s0; p1a[r] = s1;
        }
        // reduce across the 16 lanes of each half-wave
        for (int msk = 1; msk < 16; msk <<= 1)
            for (int r = 0; r < 8; ++r) {
                p0a[r] += __shfl_xor(p0a[r], msk, 32);
                p1a[r] += __shfl_xor(p1a[r], msk, 32);
            }
        if ((lane & 15) == 0) {
            for (int r = 0; r < 8; ++r) {
                size_t p = p0 + r + hi * 8;
                out[p * 2 + 0] = p0a[r] + b20;
                out[p * 2 + 1] = p1a[r] + b21;
            }
        }
    }
}

// ---------------- host-side orchestration ----------------
extern "C" void kernel_launch(void* const* d_in, const int* in_sizes, int n_in,
                              void* d_out, int out_size, void* d_ws, size_t ws_size,
                              hipStream_t stream) {
    (void)in_sizes; (void)n_in; (void)out_size; (void)ws_size;
    const float* h    = (const float*)d_in[0];
    const float* xg   = (const float*)d_in[1];
    const float* fc0w = (const float*)d_in[2];
    const float* fc0b = (const float*)d_in[3];
    const float* scwr[3] = {(const float*)d_in[4], (const float*)d_in[6], (const float*)d_in[8]};
    const float* scwi[3] = {(const float*)d_in[5], (const float*)d_in[7], (const float*)d_in[9]};
    const float* cww[3]  = {(const float*)d_in[10], (const float*)d_in[12], (const float*)d_in[14]};
    const float* cwb[3]  = {(const float*)d_in[11], (const float*)d_in[13], (const float*)d_in[15]};
    const float* fc1w = (const float*)d_in[16];
    const float* fc1b = (const float*)d_in[17];
    const float* fc2w = (const float*)d_in[18];
    const float* fc2b = (const float*)d_in[19];
    float* out = (float*)d_out;

    float* ws = (float*)d_ws;
    size_t off = 0;
    float* v0   = ws + off; off += (size_t)PTS * WCH;                    // 168 MB
    float* v1   = ws + off; off += (size_t)PTS * WCH;                    // 168 MB
    float* cbuf = ws + off; off += (size_t)PTS * WCH;                    // 168 MB
    float2* Vt  = (float2*)(ws + off); off += (size_t)LINES * KT * WCH * 2;          // 67 MB (also Wy)
    float2* Vy  = (float2*)(ws + off); off += (size_t)BN * NX * KYM * KT * WCH * 2;  // 25 MB (also Wx)
    float2* Vx  = (float2*)(ws + off); off += (size_t)BN * KXM * KYM * KT * WCH * 2; // 9.4 MB
    float2* Vm  = (float2*)(ws + off); off += (size_t)BN * KXM * KYM * KT * WCH * 2; // 9.4 MB
    float2* Wc  = (float2*)(ws + off); off += (size_t)4 * 12 * 12 * KT * WCH * WCH * 2; // 38 MB

    dim3 b256(256);
    k_fc0<<<(PTS * WCH) / 256, b256, 0, stream>>>(h, xg, fc0w, fc0b, v0);

    float* vin = v0;
    float* vout = v1;
    for (int l = 0; l < 3; ++l) {
        k_repack<<<(4 * 12 * 12 * KT * WCH * WCH) / 256, b256, 0, stream>>>(scwr[l], scwi[l], Wc);
        k_conv1x1<<<1024, b256, 0, stream>>>(vin, cww[l], cwb[l], cbuf, PTS / 16);
        k_dft_t<<<LINES, b256, 0, stream>>>(vin, Vt);
        k_dft_y<<<BN * NX * KYM, b256, 0, stream>>>(Vt, Vy);
        k_dft_x<<<BN * KXM * KYM, b256, 0, stream>>>(Vy, Vx);
        k_specmul<<<BN * KXM * KYM, b256, 0, stream>>>(Vx, Wc, Vm);
        k_idft_x<<<BN * NX * KYM, b256, 0, stream>>>(Vm, Vy /* reused as Wx */);
        k_idft_y<<<LINES, b256, 0, stream>>>(Vy, Vt /* reused as Wy */);
        k_combine<<<LINES, b256, 0, stream>>>(Vt, cbuf, vout, (l < 2) ? 1 : 0);
        float* tmp = vin; vin = vout; vout = tmp;
    }
    k_fc12<<<1024, b256, 0, stream>>>(vin, fc1w, fc1b, fc2w, fc2b, out, PTS / 16);
}